// NonLocalBlock2D_12446815224184
// MI455X (gfx1250) — compile-verified
//
#include <hip/hip_runtime.h>
#include <math.h>

// NonLocalBlock2D for MI455X (gfx1250), wave32, f32 WMMA (V_WMMA_F32_16X16X4_F32).
// B=4, C_IN=256, C_INTER=128, N=64*64=4096.
//
// Workspace tensors use a row-pair-interleaved layout so every WMMA fragment
// (two elements of adjacent k-rows in one lane) is a single coalesced b64 load:
//   theta2/phi2/Y2 : element (c, n) at [((c>>1)*N + n)*2 + (c&1)]
//   gT2            : element (j, c) at [((j>>1)*CI + c)*2 + (j&1)]
// Total workspace: 4 x 8 MB + stats = ~33.7 MB (fits easily in 192 MB L2).

#define BATCH 4
#define CIN   256
#define CI    128
#define NSP   4096

typedef __attribute__((ext_vector_type(2))) float v2f;
typedef __attribute__((ext_vector_type(8))) float v8f;

__device__ __forceinline__ v8f wmma4(v2f a, v2f b, v8f c) {
  // D = A(16x4 f32) * B(4x16 f32) + C(16x16 f32)
  return __builtin_amdgcn_wmma_f32_16x16x4_f32(false, a, false, b, (short)0, c,
                                               false, false);
}

// Online (max, sum-exp) update for the 8 column-values a lane holds; half-wave
// shuffle combines the two row-groups of each column.
__device__ __forceinline__ void online_update(const v8f& d, float& m, float& z) {
  float tmax = d[0];
  #pragma unroll
  for (int v = 1; v < 8; ++v) tmax = fmaxf(tmax, d[v]);
  tmax = fmaxf(tmax, __shfl_xor(tmax, 16, 32));
  const float nm = fmaxf(m, tmax);
  float se = 0.0f;
  #pragma unroll
  for (int v = 0; v < 8; ++v) se += __expf(d[v] - nm);
  se += __shfl_xor(se, 16, 32);
  z = z * __expf(m - nm) + se;
  m = nm;
}

// ---------------------------------------------------------------------------
// Kernel 1: theta/phi/g projections.  out[o,n] = sum_c W[o,c]*X[c,n] + b[o]
// grid.z = batch*3 + proj.  theta/phi stored pair-interleaved; g stored
// transposed + pair-interleaved for the attention kernel's B fragments.
// ---------------------------------------------------------------------------
__global__ __launch_bounds__(128)
void proj_kernel(const float* __restrict__ x,
                 const float* __restrict__ w_t, const float* __restrict__ b_t,
                 const float* __restrict__ w_p, const float* __restrict__ b_p,
                 const float* __restrict__ w_g, const float* __restrict__ b_g,
                 float* __restrict__ theta2, float* __restrict__ phi2,
                 float* __restrict__ gT2)
{
  const int z = blockIdx.z;
  const int b = z / 3, p = z % 3;
  const float* W; const float* bias;
  if (p == 0)      { W = w_t; bias = b_t; }
  else if (p == 1) { W = w_p; bias = b_p; }
  else             { W = w_g; bias = b_g; }

  const int wv   = (blockIdx.x * blockDim.x + threadIdx.x) >> 5; // 0..511
  const int lane = threadIdx.x & 31;
  const int lm = lane & 15, lh = lane >> 4;
  const int o0 = (wv & 7) * 16;     // O = 128 -> 8 tiles
  const int n0 = (wv >> 3) * 64;    // 64 groups of 64 columns

  const float* X = x + (size_t)b * CIN * NSP;

  v8f acc[4];
  #pragma unroll
  for (int s = 0; s < 4; ++s) acc[s] = (v8f){0,0,0,0,0,0,0,0};

  for (int c = 0; c < CIN; c += 4) {
    v2f a;
    const float* wr = W + (size_t)(o0 + lm) * CIN + c + lh * 2;
    a.x = wr[0]; a.y = wr[1];
    const float* xr = X + (size_t)(c + lh * 2) * NSP + n0 + lm;
    #pragma unroll
    for (int s = 0; s < 4; ++s) {
      v2f bb;
      bb.x = xr[s * 16];
      bb.y = xr[NSP + s * 16];
      acc[s] = wmma4(a, bb, acc[s]);
    }
  }

  float bv[8];
  #pragma unroll
  for (int v = 0; v < 8; ++v) bv[v] = bias[o0 + lh * 8 + v];

  if (p < 2) {
    float* out = (p == 0 ? theta2 : phi2) + (size_t)b * CI * NSP;
    #pragma unroll
    for (int s = 0; s < 4; ++s) {
      #pragma unroll
      for (int v = 0; v < 8; v += 2) {  // pair-interleaved b64 stores
        v2f t;
        t.x = acc[s][v] + bv[v];
        t.y = acc[s][v + 1] + bv[v + 1];
        *(v2f*)(out + ((size_t)((o0 + lh * 8 + v) >> 1) * NSP + n0 + s * 16 + lm) * 2) = t;
      }
    }
  } else {
    float* out = gT2 + (size_t)b * NSP * CI;
    #pragma unroll
    for (int s = 0; s < 4; ++s) {
      const int n = n0 + s * 16 + lm;
      float* q = out + ((size_t)(n >> 1) * CI + o0 + lh * 8) * 2 + (n & 1);
      #pragma unroll
      for (int v = 0; v < 8; ++v) q[v * 2] = acc[s][v] + bv[v];
    }
  }
}

// ---------------------------------------------------------------------------
// Kernel 2: per-column softmax stats (softmax over axis i for each column j).
// Each wave owns 32 columns (two 16-wide j-tiles; phi tiles in 128 VGPRs) and
// streams theta over all i: one b64 A-fragment feeds two WMMAs per k-step.
// ---------------------------------------------------------------------------
__global__ __launch_bounds__(256)
void colstats_kernel(const float* __restrict__ theta2, const float* __restrict__ phi2,
                     float* __restrict__ mbuf, float* __restrict__ zinv)
{
  const int b    = blockIdx.z;
  const int wv   = blockIdx.x * 8 + (threadIdx.x >> 5);  // 0..127
  const int lane = threadIdx.x & 31;
  const int lm = lane & 15, lh = lane >> 4;
  const int j0 = wv * 32;

  const float* th = theta2 + (size_t)b * CI * NSP;
  const float* ph = phi2   + (size_t)b * CI * NSP;

  v2f bf0[32], bf1[32];
  #pragma unroll
  for (int s = 0; s < 32; ++s) {
    const int k2 = 2 * s + lh;   // row-pair index (k = 4s + lh*2)
    bf0[s] = *(const v2f*)(ph + ((size_t)k2 * NSP + j0 + lm) * 2);
    bf1[s] = *(const v2f*)(ph + ((size_t)k2 * NSP + j0 + 16 + lm) * 2);
  }

  float m0 = -1.0e30f, z0 = 0.0f, m1 = -1.0e30f, z1 = 0.0f;
  for (int i0 = 0; i0 < NSP; i0 += 16) {
    v8f d0 = (v8f){0,0,0,0,0,0,0,0};
    v8f d1 = (v8f){0,0,0,0,0,0,0,0};
    #pragma unroll
    for (int s = 0; s < 32; ++s) {
      const int k2 = 2 * s + lh;
      const v2f a = *(const v2f*)(th + ((size_t)k2 * NSP + i0 + lm) * 2);
      d0 = wmma4(a, bf0[s], d0);
      d1 = wmma4(a, bf1[s], d1);
    }
    online_update(d0, m0, z0);
    online_update(d1, m1, z1);
  }
  if (lh == 0) {
    mbuf[(size_t)b * NSP + j0 + lm]      = m0;
    zinv[(size_t)b * NSP + j0 + lm]      = 1.0f / z0;
    mbuf[(size_t)b * NSP + j0 + 16 + lm] = m1;
    zinv[(size_t)b * NSP + j0 + 16 + lm] = 1.0f / z1;
  }
}

// ---------------------------------------------------------------------------
// Kernel 3: Y[c,i] = sum_j exp(f_ij - m_j)*zinv_j * g[c,j].
// Each wave owns 16 rows i (theta fragments resident in 64 VGPRs); per j-tile:
// 32 WMMAs rebuild the f-tile (b64 phi fragments), exp applied, P tile bounced
// through a padded per-wave LDS slice (b64 refragment reads), then 32 WMMAs
// accumulate Y^T = P * gT with b64 gT fragments.
// ---------------------------------------------------------------------------
__global__ __launch_bounds__(256)
void attn_y_kernel(const float* __restrict__ theta2, const float* __restrict__ phi2,
                   const float* __restrict__ gT2,
                   const float* __restrict__ mbuf, const float* __restrict__ zinv,
                   float* __restrict__ Y2)
{
  __shared__ __align__(16) float lds_p[8 * 16 * 18];  // 16x16 P tile/wave, pad 2
  const int b     = blockIdx.z;
  const int wslot = threadIdx.x >> 5;
  const int wv    = blockIdx.x * 8 + wslot;
  const int lane  = threadIdx.x & 31;
  const int lm = lane & 15, lh = lane >> 4;
  const int i0 = wv * 16;
  float* P = lds_p + wslot * 16 * 18;

  const float* th = theta2 + (size_t)b * CI * NSP;
  const float* ph = phi2   + (size_t)b * CI * NSP;
  const float* gt = gT2    + (size_t)b * NSP * CI;
  const float* mb = mbuf + (size_t)b * NSP;
  const float* zb = zinv + (size_t)b * NSP;

  v2f afrag[32];
  #pragma unroll
  for (int s = 0; s < 32; ++s)
    afrag[s] = *(const v2f*)(th + ((size_t)(2 * s + lh) * NSP + i0 + lm) * 2);

  v8f yacc[8];
  #pragma unroll
  for (int t = 0; t < 8; ++t) yacc[t] = (v8f){0,0,0,0,0,0,0,0};

  for (int j0 = 0; j0 < NSP; j0 += 16) {
    v8f d = (v8f){0,0,0,0,0,0,0,0};
    #pragma unroll
    for (int s = 0; s < 32; ++s) {
      const v2f bb = *(const v2f*)(ph + ((size_t)(2 * s + lh) * NSP + j0 + lm) * 2);
      d = wmma4(afrag[s], bb, d);
    }
    const float mj = mb[j0 + lm];
    const float zi = zb[j0 + lm];
    // D layout: row i = lh*8+v, col j = lm  -> store P row-major (padded)
    #pragma unroll
    for (int v = 0; v < 8; ++v)
      P[(lh * 8 + v) * 18 + lm] = __expf(d[v] - mj) * zi;
    asm volatile("s_wait_dscnt 0" ::: "memory");  // intra-wave LDS RAW fence
    // Re-fragment P as WMMA A operand: A[m = i = lane%16, k = jlocal] (b64)
    v2f pf[4];
    #pragma unroll
    for (int s4 = 0; s4 < 4; ++s4)
      pf[s4] = *(const v2f*)(P + lm * 18 + 4 * s4 + lh * 2);
    // Y^T[i, c] += P[i, j] * gT[j, c]
    #pragma unroll
    for (int ct = 0; ct < 8; ++ct) {
      #pragma unroll
      for (int s4 = 0; s4 < 4; ++s4) {
        const v2f gb = *(const v2f*)(
            gt + ((size_t)((j0 >> 1) + 2 * s4 + lh) * CI + ct * 16 + lm) * 2);
        yacc[ct] = wmma4(pf[s4], gb, yacc[ct]);
      }
    }
  }
  // yacc[ct]: D[m = i local = lh*8+v, n = c local = lm] -> Y2 pair-interleaved
  float* Yb = Y2 + (size_t)b * CI * NSP;
  #pragma unroll
  for (int ct = 0; ct < 8; ++ct) {
    const int c = ct * 16 + lm;
    float* q = Yb + ((size_t)(c >> 1) * NSP + i0 + lh * 8) * 2 + (c & 1);
    #pragma unroll
    for (int v = 0; v < 8; ++v) q[v * 2] = yacc[ct][v];
  }
}

// ---------------------------------------------------------------------------
// Kernel 4: out[o,n] = sum_c W_w[o,c]*Y[c,n] + W_b[o] + x[o,n]
// Y read as b64 pair-interleaved fragments (Y is read 16x, written 1x).
// ---------------------------------------------------------------------------
__global__ __launch_bounds__(128)
void out_kernel(const float* __restrict__ Y2, const float* __restrict__ Ww,
                const float* __restrict__ Wb, const float* __restrict__ x,
                float* __restrict__ out)
{
  const int b    = blockIdx.z;
  const int wv   = (blockIdx.x * blockDim.x + threadIdx.x) >> 5; // 0..1023
  const int lane = threadIdx.x & 31;
  const int lm = lane & 15, lh = lane >> 4;
  const int o0 = (wv & 15) * 16;   // O = 256 -> 16 tiles
  const int n0 = (wv >> 4) * 64;

  const float* Yb = Y2 + (size_t)b * CI * NSP;
  const float* xb = x + (size_t)b * CIN * NSP;
  float* ob = out + (size_t)b * CIN * NSP;

  v8f acc[4];
  #pragma unroll
  for (int s = 0; s < 4; ++s) acc[s] = (v8f){0,0,0,0,0,0,0,0};

  for (int c = 0; c < CI; c += 4) {
    v2f a;
    const float* wr = Ww + (size_t)(o0 + lm) * CI + c + lh * 2;
    a.x = wr[0]; a.y = wr[1];
    const float* yr = Yb + ((size_t)((c >> 1) + lh) * NSP + n0 + lm) * 2;
    #pragma unroll
    for (int s = 0; s < 4; ++s) {
      const v2f bb = *(const v2f*)(yr + s * 32);
      acc[s] = wmma4(a, bb, acc[s]);
    }
  }
  #pragma unroll
  for (int s = 0; s < 4; ++s)
    #pragma unroll
    for (int v = 0; v < 8; ++v) {
      const size_t idx = (size_t)(o0 + lh * 8 + v) * NSP + n0 + s * 16 + lm;
      ob[idx] = acc[s][v] + Wb[o0 + lh * 8 + v] + xb[idx];
    }
}

// ---------------------------------------------------------------------------
extern "C" void kernel_launch(void* const* d_in, const int* in_sizes, int n_in,
                              void* d_out, int out_size, void* d_ws, size_t ws_size,
                              hipStream_t stream)
{
  const float* x       = (const float*)d_in[0];
  const float* theta_w = (const float*)d_in[1];
  const float* theta_b = (const float*)d_in[2];
  const float* phi_w   = (const float*)d_in[3];
  const float* phi_b   = (const float*)d_in[4];
  const float* g_w     = (const float*)d_in[5];
  const float* g_b     = (const float*)d_in[6];
  const float* W_w     = (const float*)d_in[7];
  const float* W_b     = (const float*)d_in[8];
  float* out = (float*)d_out;

  float* ws = (float*)d_ws;
  const size_t per = (size_t)BATCH * CI * NSP;   // 2,097,152 floats
  float* theta2 = ws;
  float* phi2   = theta2 + per;
  float* gT2    = phi2 + per;
  float* Y2     = gT2 + per;
  float* mbuf   = Y2 + per;
  float* zbuf   = mbuf + (size_t)BATCH * NSP;    // needs ~33.7 MB of ws total

  proj_kernel<<<dim3(128, 1, BATCH * 3), 128, 0, stream>>>(
      x, theta_w, theta_b, phi_w, phi_b, g_w, g_b, theta2, phi2, gT2);
  colstats_kernel<<<dim3(16, 1, BATCH), 256, 0, stream>>>(theta2, phi2, mbuf, zbuf);
  attn_y_kernel<<<dim3(32, 1, BATCH), 256, 0, stream>>>(theta2, phi2, gT2, mbuf, zbuf, Y2);
  out_kernel<<<dim3(256, 1, BATCH), 128, 0, stream>>>(Y2, W_w, W_b, x, out);
}